// GraphSageLayer_86784109183566
// MI455X (gfx1250) — compile-verified
//
#include <hip/hip_runtime.h>

typedef __attribute__((ext_vector_type(2))) float v2f;
typedef __attribute__((ext_vector_type(8))) float v8f;

#define N_NODES   100000
#define N_EDGES   625000
#define FEAT      128
#define TWOF      256
#define N_DISEASE 50000
#define SLOPE     0.2f
#define DEG_EPS   1e-6f
#define A_STRIDE  260   // 256 + pad(4): avoids 16-way LDS bank conflicts, keeps 16B alignment

// ---------------- kernel 0: zero h_agg + deg in workspace ----------------
__global__ __launch_bounds__(256) void k_zero(float* __restrict__ ws, int n4) {
  int i = blockIdx.x * blockDim.x + threadIdx.x;
  if (i < n4) ((float4*)ws)[i] = make_float4(0.f, 0.f, 0.f, 0.f);
}

// ---------------- kernel 1: edge scatter-add (one wave32 per edge) --------
__global__ __launch_bounds__(256) void k_scatter(const float* __restrict__ h,
                                                 const int* __restrict__ src,
                                                 const int* __restrict__ dst,
                                                 float* __restrict__ hagg,
                                                 float* __restrict__ deg) {
  int wave = (blockIdx.x * blockDim.x + threadIdx.x) >> 5;
  int lane = threadIdx.x & 31;
  if (wave >= N_EDGES) return;
  int s = src[wave];
  int d = dst[wave];
  // 32 lanes x float4 = 128 feats, fully coalesced 512B gather per edge
  float4 v = ((const float4*)(h + (size_t)s * FEAT))[lane];
  float* p = hagg + (size_t)d * FEAT + lane * 4;
  __hip_atomic_fetch_add(p + 0, v.x, __ATOMIC_RELAXED, __HIP_MEMORY_SCOPE_AGENT);
  __hip_atomic_fetch_add(p + 1, v.y, __ATOMIC_RELAXED, __HIP_MEMORY_SCOPE_AGENT);
  __hip_atomic_fetch_add(p + 2, v.z, __ATOMIC_RELAXED, __HIP_MEMORY_SCOPE_AGENT);
  __hip_atomic_fetch_add(p + 3, v.w, __ATOMIC_RELAXED, __HIP_MEMORY_SCOPE_AGENT);
  if (lane == 0)
    __hip_atomic_fetch_add(deg + d, 1.0f, __ATOMIC_RELAXED, __HIP_MEMORY_SCOPE_AGENT);
}

// ---------------- kernel 2: h_agg /= max(deg, eps), in place --------------
__global__ __launch_bounds__(256) void k_norm(float* __restrict__ hagg,
                                              const float* __restrict__ deg, int n4) {
  int i = blockIdx.x * blockDim.x + threadIdx.x;
  if (i >= n4) return;
  int node = i >> 5;                        // (i*4)/128
  float inv = 1.0f / fmaxf(deg[node], DEG_EPS);
  float4 v = ((float4*)hagg)[i];
  v.x *= inv; v.y *= inv; v.z *= inv; v.w *= inv;
  ((float4*)hagg)[i] = v;
}

// ---------------- kernel 3: fused concat-GEMM + bias + LeakyReLU ----------
// Block = 256 threads (8 wave32). Block owns 16 nodes; wave w owns output
// features [16w, 16w+16). K = 256 swept 4 at a time with V_WMMA_F32_16X16X4_F32.
__global__ __launch_bounds__(256) void k_gemm(const float* __restrict__ h,
                                              const float* __restrict__ haggn,
                                              const float* __restrict__ Wd,
                                              const float* __restrict__ bd,
                                              const float* __restrict__ Wm,
                                              const float* __restrict__ bm,
                                              float* __restrict__ out) {
  __shared__ float Ah[16 * A_STRIDE];       // 16 nodes x 256 k (padded)

  const int node0 = blockIdx.x * 16;
  const bool is_d = (node0 < N_DISEASE);    // 50000 % 16 == 0: tiles never straddle
  const float* __restrict__ W    = is_d ? Wd : Wm;
  const float* __restrict__ bias = is_d ? bd : bm;
  const int tid = threadIdx.x;

  // Stage A = concat(h, h_agg/deg) rows node0..node0+15 into LDS, float4 per thread.
  #pragma unroll
  for (int it = 0; it < 4; ++it) {
    int e = it * 1024 + tid * 4;            // flat element in 16x256 tile
    int m = e >> 8;                         // node row 0..15
    int k = e & 255;                        // concat column
    const float* srcp = (k < FEAT) ? (h     + (size_t)(node0 + m) * FEAT + k)
                                   : (haggn + (size_t)(node0 + m) * FEAT + (k - FEAT));
    *(float4*)(Ah + m * A_STRIDE + k) = *(const float4*)srcp;
  }
  __syncthreads();

  const int wave  = tid >> 5;
  const int lane  = tid & 31;
  const int col   = wave * 16 + (lane & 15); // output feature (N)
  const int mrow  = lane & 15;               // A row this lane carries
  const int khalf = (lane >> 4) * 2;         // lanes 0-15: K=0,1  lanes 16-31: K=2,3

  v8f acc = {};
  #pragma unroll 8
  for (int kk = 0; kk < TWOF; kk += 4) {
    // A fragment: 16x4 f32 — lane holds (mrow, kk+khalf) and (mrow, kk+khalf+1)
    v2f a = *(const v2f*)(Ah + mrow * A_STRIDE + kk + khalf);
    // B fragment: 4x16 f32 — row striped across lanes, mirrored half-wave K split
    v2f b;
    b.x = W[(size_t)(kk + khalf)     * FEAT + col];
    b.y = W[(size_t)(kk + khalf + 1) * FEAT + col];
    acc = __builtin_amdgcn_wmma_f32_16x16x4_f32(
        /*neg_a=*/false, a, /*neg_b=*/false, b,
        /*c_mod=*/(short)0, acc, /*reuse_a=*/false, /*reuse_b=*/false);
  }

  // Epilogue: C/D layout — VGPR r: lanes0-15 -> M=r, lanes16-31 -> M=r+8
  float bv = bias[col];
  #pragma unroll
  for (int r = 0; r < 8; ++r) {
    int m = (lane < 16) ? r : (r + 8);
    float val = acc[r] + bv;
    val = (val >= 0.f) ? val : val * SLOPE;
    out[(size_t)(node0 + m) * FEAT + col] = val;
  }
}

extern "C" void kernel_launch(void* const* d_in, const int* in_sizes, int n_in,
                              void* d_out, int out_size, void* d_ws, size_t ws_size,
                              hipStream_t stream) {
  (void)in_sizes; (void)n_in; (void)out_size; (void)ws_size;
  const float* h  = (const float*)d_in[0];
  const int*   src = (const int*)d_in[1];
  const int*   dst = (const int*)d_in[2];
  const float* Wd = (const float*)d_in[3];
  const float* bd = (const float*)d_in[4];
  const float* Wm = (const float*)d_in[5];
  const float* bm = (const float*)d_in[6];
  float* out  = (float*)d_out;
  float* hagg = (float*)d_ws;                        // [N_NODES * FEAT]
  float* deg  = hagg + (size_t)N_NODES * FEAT;       // [N_NODES]

  // ws must be re-zeroed every call (harness does not re-poison between replays)
  int zero4 = (N_NODES * FEAT + N_NODES) / 4;        // 3,225,000 float4s
  k_zero<<<(zero4 + 255) / 256, 256, 0, stream>>>((float*)d_ws, zero4);

  k_scatter<<<N_EDGES / 8, 256, 0, stream>>>(h, src, dst, hagg, deg);

  int n4 = N_NODES * FEAT / 4;
  k_norm<<<(n4 + 255) / 256, 256, 0, stream>>>(hagg, deg, n4);

  k_gemm<<<N_NODES / 16, 256, 0, stream>>>(h, hagg, Wd, bd, Wm, bm, out);
}